// zak_gpr_53412213293069
// MI455X (gfx1250) — compile-verified
//
#include <hip/hip_runtime.h>
#include <math.h>

typedef __attribute__((ext_vector_type(2))) float v2f;
typedef __attribute__((ext_vector_type(8))) float v8f;

#define NOISE   0.01f
#define TH_S    8.0f
#define TH_T    10.0f
#define LOG2PI  1.8378770664093453f
#define LOG_B   0.9189385332046727f

// ---------------------------------------------------------------------------
// Generic strided WMMA f32 GEMM:  C[m,n] = beta*C[m,n] + sum_k A[m,k]*B[k,n]
// A element at A[m*Arm + k*Ack], B element at B[k*Brk + n*Bcn], C row stride Crm.
// One wave (32 lanes) computes one 16x16 tile via V_WMMA_F32_16X16X4_F32.
// ---------------------------------------------------------------------------
__global__ __launch_bounds__(32) void gemm_wmma_f32(
    const float* __restrict__ A, const float* __restrict__ B, float* __restrict__ C,
    int M, int N, int K, int Arm, int Ack, int Brk, int Bcn, int Crm, float beta)
{
    const int ntile = N >> 4;
    const int tile  = blockIdx.x;
    const int tm    = (tile / ntile) << 4;
    const int tn    = (tile % ntile) << 4;
    const int lane  = threadIdx.x & 31;
    const int half  = lane >> 4;     // 0: K pair {0,1}, 1: K pair {2,3}
    const int l16   = lane & 15;
    const int am    = tm + l16;      // A-frag row (M)
    const int bn    = tn + l16;      // B-frag col (N)
    v8f acc = {0.f,0.f,0.f,0.f,0.f,0.f,0.f,0.f};
    for (int k0 = 0; k0 < K; k0 += 4) {
        const int ka = k0 + (half << 1);
        v2f av, bv;
        av.x = A[am * Arm + ka * Ack];
        av.y = A[am * Arm + (ka + 1) * Ack];
        bv.x = B[ka * Brk + bn * Bcn];
        bv.y = B[(ka + 1) * Brk + bn * Bcn];
        acc = __builtin_amdgcn_wmma_f32_16x16x4_f32(
                  false, av, false, bv, (short)0, acc, false, false);
    }
#pragma unroll
    for (int r = 0; r < 8; ++r) {
        const int m = tm + r + (half << 3);   // VGPR r: M=r (lanes 0-15), M=r+8 (16-31)
        const int n = tn + l16;
        float prev = (beta != 0.0f) ? C[m * Crm + n] : 0.0f;
        C[m * Crm + n] = beta * prev + acc[r];
    }
}

// ---------------------------------------------------------------------------
// Build the Kronecker factors from coordinates:
//   Ks[i,j] = exp(-(s_i-s_j)^2/8), Kt[i,j] = exp(-(t_i-t_j)^2/10),
//   G[k,p]  = exp(-(ts_p-s_k)^2/8) (64x16, cols 8..15 zero-padded)
// ---------------------------------------------------------------------------
__global__ void build_factors(const float* __restrict__ X, const float* __restrict__ Xt,
                              float* Ks, float* Kt, float* G)
{
    int idx = blockIdx.x * blockDim.x + threadIdx.x;
    if (idx < 4096) {
        int i = idx >> 6, j = idx & 63;
        float si = X[(i << 6) * 2], sj = X[(j << 6) * 2];
        float ds = si - sj;
        Ks[idx] = __expf(-ds * ds / TH_S);
        float ti = X[i * 2 + 1], tj = X[j * 2 + 1];
        float dt = ti - tj;
        Kt[idx] = __expf(-dt * dt / TH_T);
    }
    if (idx < 1024) {
        int k = idx >> 4, p = idx & 15;
        float v = 0.f;
        if (p < 8) {
            float tsp = Xt[(p << 6) * 2];
            float sk  = X[(k << 6) * 2];
            float d = tsp - sk;
            v = __expf(-d * d / TH_S);
        }
        G[idx] = v;
    }
}

// ---------------------------------------------------------------------------
// Cyclic two-sided Jacobi eigensolver for symmetric 64x64: A = V diag(lam) V^T
// One block of 64 threads; A,V live in LDS.
// ---------------------------------------------------------------------------
__global__ __launch_bounds__(64) void jacobi64(const float* __restrict__ Ain,
                                               float* U, float* lam)
{
    __shared__ float A[64][65];
    __shared__ float V[64][65];
    __shared__ float cs[2];
    int t = threadIdx.x;
    for (int j = 0; j < 64; ++j) { A[t][j] = Ain[t * 64 + j]; V[t][j] = (t == j) ? 1.f : 0.f; }
    __syncthreads();
    for (int sweep = 0; sweep < 12; ++sweep) {
        for (int p = 0; p < 63; ++p) {
            for (int q = p + 1; q < 64; ++q) {
                if (t == 0) {
                    float apq = A[p][q], app = A[p][p], aqq = A[q][q];
                    float c = 1.f, s = 0.f;
                    if (fabsf(apq) > 1e-12f) {
                        float tau = (aqq - app) / (2.f * apq);
                        float tt = (tau >= 0.f) ? 1.f / (tau + sqrtf(1.f + tau * tau))
                                                : 1.f / (tau - sqrtf(1.f + tau * tau));
                        c = rsqrtf(1.f + tt * tt);
                        s = tt * c;
                    }
                    cs[0] = c; cs[1] = s;
                }
                __syncthreads();
                float c = cs[0], s = cs[1];
                float aip = A[t][p], aiq = A[t][q];
                A[t][p] = c * aip - s * aiq;  A[t][q] = s * aip + c * aiq;
                __syncthreads();
                float api = A[p][t], aqi = A[q][t];
                A[p][t] = c * api - s * aqi;  A[q][t] = s * api + c * aqi;
                __syncthreads();
                float vip = V[t][p], viq = V[t][q];
                V[t][p] = c * vip - s * viq;  V[t][q] = s * vip + c * viq;
                __syncthreads();
            }
        }
    }
    lam[t] = A[t][t];
    for (int j = 0; j < 64; ++j) U[t * 64 + j] = V[t][j];
}

// D, t-tilde, Ht from eigvals/eigvecs:
//   invD[i,j] = 1/(a^2 ls_i lt_j + noise); tt[j,tau] = lt_j * Ut[tau,j]; Ht = tt^2
__global__ void build_eig_aux(const float* ls, const float* lt, const float* Ut,
                              const float* alpha, float* invD, float* tt, float* Ht)
{
    int idx = blockIdx.x * blockDim.x + threadIdx.x;
    if (idx >= 4096) return;
    float a = alpha[0];
    int i = idx >> 6, j = idx & 63;
    float d = a * a * ls[i] * lt[j] + NOISE;
    invD[idx] = 1.0f / d;
    float tv = lt[i] * Ut[j * 64 + i];   // tt[i][j] with i=eig-idx, j=tau
    tt[idx] = tv;
    Ht[idx] = tv * tv;
}

__global__ __launch_bounds__(64) void colsum64(const float* Ut, float* wv)
{
    int j = threadIdx.x; float s = 0.f;
    for (int tau = 0; tau < 64; ++tau) s += Ut[tau * 64 + j];
    wv[j] = s;
}

__global__ __launch_bounds__(64) void rowsum_q(const float* invD, const float* wv, float* q)
{
    int i = threadIdx.x; float s = 0.f;
    for (int j = 0; j < 64; ++j) { float wj = wv[j]; s += wj * wj * invD[i * 64 + j]; }
    q[i] = s;
}

__global__ void ew_mul(const float* x, const float* y, float* z, int n)
{
    int i = blockIdx.x * blockDim.x + threadIdx.x;
    if (i < n) z[i] = x[i] * y[i];
}

__global__ void build_E(const float* invD, const float* wv, float* E)
{
    int i = blockIdx.x * blockDim.x + threadIdx.x;
    if (i < 4096) E[i] = invD[i] * wv[i & 63];
}

__global__ void build_denom(const float* holder, const float* alpha, float* den)
{
    int n = blockIdx.x * blockDim.x + threadIdx.x;
    if (n < 512) { float a = alpha[0]; den[n] = 1.0f / (a * a) - holder[n]; }
}

// Bbig[i, p*64+tau] = gt[i,p] * T[i,tau]   (i,eig x 512 test pts)
__global__ void scale_bbig(const float* gt, const float* T, float* Bbig)
{
    int idx = blockIdx.x * blockDim.x + threadIdx.x;
    if (idx >= 64 * 512) return;
    int i = idx >> 9, n = idx & 511, p = n >> 6, tau = n & 63;
    Bbig[idx] = gt[i * 16 + p] * T[i * 64 + tau];
}

__global__ void build_Pd(const float* P, const float* den, float* Pd)
{
    int idx = blockIdx.x * blockDim.x + threadIdx.x;
    if (idx < 64 * 512) Pd[idx] = P[idx] / den[idx & 511];
}

__global__ void init_A(const float* Sblk, float* Amat)
{
    int idx = blockIdx.x * blockDim.x + threadIdx.x;
    if (idx < 4096) Amat[idx] = Sblk[idx] + (((idx >> 6) == (idx & 63)) ? 1.0f : 0.0f);
}

__global__ void build_B2(const float* q, const float* Us, float* B2)
{
    int idx = blockIdx.x * blockDim.x + threadIdx.x;
    if (idx >= 4096) return;
    int i = idx >> 6, k = idx & 63;
    B2[idx] = q[i] * Us[k * 64 + i];
}

__global__ void build_Rtil(const float* Ytil, const float* bs, const float* wv, float* Rtil)
{
    int idx = blockIdx.x * blockDim.x + threadIdx.x;
    if (idx >= 4096) return;
    Rtil[idx] = Ytil[idx] - bs[idx >> 6] * wv[idx & 63];
}

__global__ __launch_bounds__(64) void matvec64(const float* Mx, const float* v,
                                               float* out, int transpose)
{
    int t = threadIdx.x; float s = 0.f;
    if (!transpose) for (int j = 0; j < 64; ++j) s += Mx[t * 64 + j] * v[j];
    else            for (int j = 0; j < 64; ++j) s += Mx[j * 64 + t] * v[j];
    out[t] = s;
}

__global__ __launch_bounds__(64) void build_Cvec(const float* Sy1, const float* P,
                                                 const float* spad, const float* Yt,
                                                 const float* den, float* Cv)
{
    int t = threadIdx.x;
    float acc = Sy1[t];
    for (int n = 0; n < 512; ++n)
        acc += P[t * 512 + n] * (spad[n] - Yt[n]) / den[n];
    Cv[t] = acc;
}

// Gaussian elimination (SPD, no pivot) for 64x64 A b = C
__global__ __launch_bounds__(64) void solve64(const float* Ain, const float* Cin, float* b)
{
    __shared__ float A[64][65];
    __shared__ float x[64];
    int t = threadIdx.x;
    for (int j = 0; j < 64; ++j) A[t][j] = Ain[t * 64 + j];
    x[t] = Cin[t];
    __syncthreads();
    for (int k = 0; k < 64; ++k) {
        if (t > k) {
            float f = A[t][k] / A[k][k];
            for (int j = k; j < 64; ++j) A[t][j] -= f * A[k][j];
            x[t] -= f * x[k];
        }
        __syncthreads();
    }
    if (t == 0) {
        for (int k = 63; k >= 0; --k) {
            float s = x[k];
            for (int j = k + 1; j < 64; ++j) s -= A[k][j] * x[j];
            x[k] = s / A[k][k];
        }
    }
    __syncthreads();
    b[t] = x[t];
}

// --- fixed-order reductions (deterministic; no atomics) ---
__global__ __launch_bounds__(256) void reduce_chunk1(const float* invD, float* out)
{   // 0.5 * sum log D = 0.5 * sum(-log invD)
    __shared__ float sh[256];
    int t = threadIdx.x; float s = 0.f;
    for (int i = t; i < 4096; i += 256) s += -__logf(invD[i]);
    sh[t] = s; __syncthreads();
    for (int o = 128; o > 0; o >>= 1) { if (t < o) sh[t] += sh[t + o]; __syncthreads(); }
    if (t == 0) out[0] = 0.5f * sh[0];
}

__global__ __launch_bounds__(256) void reduce_chunk2(const float* R, const float* invD, float* out)
{   // -0.5 * sum R^2 * invD
    __shared__ float sh[256];
    int t = threadIdx.x; float s = 0.f;
    for (int i = t; i < 4096; i += 256) { float r = R[i]; s += r * r * invD[i]; }
    sh[t] = s; __syncthreads();
    for (int o = 128; o > 0; o >>= 1) { if (t < o) sh[t] += sh[t + o]; __syncthreads(); }
    if (t == 0) out[0] = -0.5f * sh[0];
}

__global__ __launch_bounds__(64) void chunk3_kernel(const float* b, const float* alpha, float* out)
{
    __shared__ float sh[64];
    int t = threadIdx.x;
    float bk = b[t];
    sh[t] = -0.5f * bk * bk + LOG_B;
    __syncthreads();
    for (int o = 32; o > 0; o >>= 1) { if (t < o) sh[t] += sh[t + o]; __syncthreads(); }
    if (t == 0) {
        float a = alpha[0];
        float prob_a = -0.5f * (a - 1.0f) * (a - 1.0f) / 0.25f + __logf(0.25f * 2.5066282746f);
        out[0] = -32.0f * LOG2PI + prob_a + sh[0] / 64.0f;
    }
}

__global__ __launch_bounds__(256) void reduce_chunk4(const float* den, const float* mu,
                                                     const float* Yt, float* out)
{
    __shared__ float sh[256];
    int t = threadIdx.x; float s = 0.f;
    for (int n = t; n < 512; n += 256) {
        float v = fabsf(den[n]);
        float d = Yt[n] - mu[n];
        s += -__logf(v) - d * d / v - LOG2PI;
    }
    sh[t] = s; __syncthreads();
    for (int o = 128; o > 0; o >>= 1) { if (t < o) sh[t] += sh[t + o]; __syncthreads(); }
    if (t == 0) out[0] = 0.5f * sh[0];
}

__global__ void final_sum(const float* scal, float* out)
{
    if (threadIdx.x == 0 && blockIdx.x == 0)
        out[0] = scal[0] + scal[1] + scal[2] + scal[3];
}

// ---------------------------------------------------------------------------
// Workspace layout (float offsets)
// ---------------------------------------------------------------------------
enum : size_t {
    O_KS = 0,       O_KT = 4096,    O_G  = 8192,    O_US = 9216,   O_UT = 13312,
    O_LS = 17408,   O_LT = 17472,   O_INVD = 17536, O_TT = 21632,  O_HT = 25728,
    O_GT = 29824,   O_HS = 30848,   O_WV = 31872,   O_E  = 31936,  O_T  = 36032,
    O_T1 = 40128,   O_HOLD = 44224, O_DEN = 45248,  O_BBIG = 45760, O_P = 78528,
    O_PD = 111296,  O_T2 = 144064,  O_YTIL = 148160, O_F = 152256, O_T3 = 156352,
    O_SPAD = 160448, O_Q = 161472,  O_B2 = 161536,  O_SBLK = 165632, O_AMAT = 169728,
    O_CV = 173824,  O_BV = 173888,  O_BS = 173952,  O_FV = 174016, O_SY1 = 174080,
    O_RTIL = 174144, O_F2 = 178240, O_T5 = 182336,  O_MU = 186432, O_SCAL = 187456,
    O_END = 187472
};

static inline void gemmL(hipStream_t st, const float* A, const float* B, float* C,
                         int M, int N, int K, int Arm, int Ack, int Brk, int Bcn,
                         int Crm, float beta)
{
    int blocks = (M / 16) * (N / 16);
    gemm_wmma_f32<<<blocks, 32, 0, st>>>(A, B, C, M, N, K, Arm, Ack, Brk, Bcn, Crm, beta);
}

extern "C" void kernel_launch(void* const* d_in, const int* in_sizes, int n_in,
                              void* d_out, int out_size, void* d_ws, size_t ws_size,
                              hipStream_t stream)
{
    const float* X     = (const float*)d_in[0];   // [4096,2]
    const float* Y     = (const float*)d_in[1];   // [4096]
    const float* alpha = (const float*)d_in[3];   // [1]
    const float* Xt    = (const float*)d_in[4];   // [512,2]
    const float* Yt    = (const float*)d_in[5];   // [512]
    float* out = (float*)d_out;
    float* W = (float*)d_ws;

    float *Ks = W+O_KS, *Kt = W+O_KT, *G = W+O_G, *Us = W+O_US, *Ut = W+O_UT;
    float *ls = W+O_LS, *lt = W+O_LT, *invD = W+O_INVD, *tt = W+O_TT, *Ht = W+O_HT;
    float *gt = W+O_GT, *Hs = W+O_HS, *wv = W+O_WV, *E = W+O_E, *T = W+O_T;
    float *T1 = W+O_T1, *hold = W+O_HOLD, *den = W+O_DEN, *Bbig = W+O_BBIG, *P = W+O_P;
    float *Pd = W+O_PD, *t2 = W+O_T2, *Ytil = W+O_YTIL, *F = W+O_F, *t3 = W+O_T3;
    float *spad = W+O_SPAD, *q = W+O_Q, *B2 = W+O_B2, *Sblk = W+O_SBLK, *Amat = W+O_AMAT;
    float *Cv = W+O_CV, *bv = W+O_BV, *bs = W+O_BS, *fv = W+O_FV, *Sy1 = W+O_SY1;
    float *Rtil = W+O_RTIL, *F2 = W+O_F2, *t5 = W+O_T5, *mu = W+O_MU, *scal = W+O_SCAL;

    // 1) Kronecker factors and eigendecompositions (64x64 each)
    build_factors<<<16, 256, 0, stream>>>(X, Xt, Ks, Kt, G);
    jacobi64<<<1, 64, 0, stream>>>(Ks, Us, ls);
    jacobi64<<<1, 64, 0, stream>>>(Kt, Ut, lt);
    colsum64<<<1, 64, 0, stream>>>(Ut, wv);
    build_eig_aux<<<16, 256, 0, stream>>>(ls, lt, Ut, alpha, invD, tt, Ht);

    // 2) gt = Us^T G  (64x16);  Hs = gt^2
    gemmL(stream, Us, G, gt, 64, 16, 64, /*A^T*/1, 64, 16, 1, 16, 0.f);
    ew_mul<<<4, 256, 0, stream>>>(gt, gt, Hs, 1024);

    // 3) holder = Hs^T (invD Ht): T1 = invD@Ht; holder(16x64) = Hs^T@T1
    gemmL(stream, invD, Ht, T1, 64, 64, 64, 64, 1, 64, 1, 64, 0.f);
    gemmL(stream, Hs, T1, hold, 16, 64, 64, /*A^T*/1, 16, 64, 1, 64, 0.f);
    build_denom<<<2, 256, 0, stream>>>(hold, alpha, den);

    // 4) P = Us @ (g~ o (E@tt)) : E=invD*w; T=E@tt; Bbig; P(64x512)
    build_E<<<16, 256, 0, stream>>>(invD, wv, E);
    gemmL(stream, E, tt, T, 64, 64, 64, 64, 1, 64, 1, 64, 0.f);
    scale_bbig<<<128, 256, 0, stream>>>(gt, T, Bbig);
    gemmL(stream, Us, Bbig, P, 64, 512, 64, 64, 1, 512, 1, 512, 0.f);

    // 5) Ytil = Us^T Ym Ut ; F = Ytil*invD ; s = gt^T (F@tt)
    gemmL(stream, Us, Y, t2, 64, 64, 64, /*A^T*/1, 64, 64, 1, 64, 0.f);
    gemmL(stream, t2, Ut, Ytil, 64, 64, 64, 64, 1, 64, 1, 64, 0.f);
    ew_mul<<<16, 256, 0, stream>>>(Ytil, invD, F, 4096);
    gemmL(stream, F, tt, t3, 64, 64, 64, 64, 1, 64, 1, 64, 0.f);
    gemmL(stream, gt, t3, spad, 16, 64, 64, /*A^T*/1, 16, 64, 1, 64, 0.f);

    // 6) Sblk = Us diag(q) Us^T
    rowsum_q<<<1, 64, 0, stream>>>(invD, wv, q);
    build_B2<<<16, 256, 0, stream>>>(q, Us, B2);
    gemmL(stream, Us, B2, Sblk, 64, 64, 64, 64, 1, 64, 1, 64, 0.f);

    // 7) A = Pd @ P^T + Sblk + I ; solve A b = C
    init_A<<<16, 256, 0, stream>>>(Sblk, Amat);
    build_Pd<<<128, 256, 0, stream>>>(P, den, Pd);
    gemmL(stream, Pd, P, Amat, 64, 64, 512, 512, 1, /*B^T*/1, 512, 64, 1.f);
    matvec64<<<1, 64, 0, stream>>>(F, wv, fv, 0);     // fv = F @ w
    matvec64<<<1, 64, 0, stream>>>(Us, fv, Sy1, 0);   // Sy rowsums = Us @ fv
    build_Cvec<<<1, 64, 0, stream>>>(Sy1, P, spad, Yt, den, Cv);
    solve64<<<1, 64, 0, stream>>>(Amat, Cv, bv);

    // 8) residual transform: Rtil = Ytil - (Us^T b) w^T ; mu = gt^T ((Rtil*invD)@tt)
    matvec64<<<1, 64, 0, stream>>>(Us, bv, bs, 1);    // bs = Us^T b
    build_Rtil<<<16, 256, 0, stream>>>(Ytil, bs, wv, Rtil);
    ew_mul<<<16, 256, 0, stream>>>(Rtil, invD, F2, 4096);
    gemmL(stream, F2, tt, t5, 64, 64, 64, 64, 1, 64, 1, 64, 0.f);
    gemmL(stream, gt, t5, mu, 16, 64, 64, /*A^T*/1, 16, 64, 1, 64, 0.f);

    // 9) scalar chunks and final sum
    reduce_chunk1<<<1, 256, 0, stream>>>(invD, scal + 0);
    reduce_chunk2<<<1, 256, 0, stream>>>(Rtil, invD, scal + 1);
    chunk3_kernel<<<1, 64, 0, stream>>>(bv, alpha, scal + 2);
    reduce_chunk4<<<1, 256, 0, stream>>>(den, mu, Yt, scal + 3);
    final_sum<<<1, 1, 0, stream>>>(scal, out);

    (void)in_sizes; (void)n_in; (void)out_size; (void)ws_size;
}